// RTL_56710748176582
// MI455X (gfx1250) — compile-verified
//
#include <hip/hip_runtime.h>
#include <stdint.h>

#define BATCH 2048
#define NFEAT 128
#define NLAT  1024
#define NVERT 81
#define KPAD  88          // 81 padded to 88 floats (352B, 16B-aligned rows)

// ---------------------------------------------------------------------------
// CDNA5 async global->LDS copy (ASYNCcnt-tracked, per CDNA5 ISA)
// vdst = VGPR with LDS byte offset, vaddr = VGPR pair with 64-bit global VA.
// ---------------------------------------------------------------------------
__device__ __forceinline__ void async_g2l_b128(uint32_t lds_byte_off,
                                               const void* gaddr) {
    unsigned long long g = (unsigned long long)(uintptr_t)gaddr;
    asm volatile("global_load_async_to_lds_b128 %0, %1, off"
                 :: "v"(lds_byte_off), "v"(g) : "memory");
}
__device__ __forceinline__ void wait_asynccnt0() {
    asm volatile("s_wait_asynccnt 0" ::: "memory");
}

// ---------------------------------------------------------------------------
// Prep 1: xT[f][b] = clip(x[b][f], 0, 2)   (LDS-tiled transpose, both sides coalesced)
// grid (NFEAT/32, BATCH/32), block 256
// ---------------------------------------------------------------------------
__global__ __launch_bounds__(256) void RTL_xpose_clip(const float* __restrict__ x,
                                                      float* __restrict__ xT) {
    __shared__ float tile[32][33];
    const int f0 = blockIdx.x * 32, b0 = blockIdx.y * 32;
    const int tj = threadIdx.x & 31, ti = threadIdx.x >> 5;
#pragma unroll
    for (int g = 0; g < 4; ++g) {
        int i = ti + g * 8;                       // b offset
        tile[i][tj] = x[(size_t)(b0 + i) * NFEAT + (f0 + tj)];
    }
    __syncthreads();
#pragma unroll
    for (int g = 0; g < 4; ++g) {
        int i = ti + g * 8;                       // f offset
        float v = tile[tj][i];
        v = fminf(fmaxf(v, 0.0f), 2.0f);
        xT[(size_t)(f0 + i) * BATCH + (b0 + tj)] = v;
    }
}

// ---------------------------------------------------------------------------
// Prep 2: Kt[l][v] = kernel[v][l], rows padded to KPAD. grid NLAT/256, block 256
// ---------------------------------------------------------------------------
__global__ __launch_bounds__(256) void RTL_kpose(const float* __restrict__ kern,
                                                 float* __restrict__ Kt) {
    const int l = blockIdx.x * 256 + threadIdx.x;
#pragma unroll 1
    for (int v = 0; v < NVERT; ++v)               // reads coalesced across lanes
        Kt[(size_t)l * KPAD + v] = kern[(size_t)v * NLAT + l];
}

// ---------------------------------------------------------------------------
// Main: block = 32 lattices x 32 batches, 256 threads (8 waves x 4 passes).
// grid (BATCH/32, NLAT/32)
// ---------------------------------------------------------------------------
__global__ __launch_bounds__(256) void RTL_main(const float* __restrict__ xT,
                                                const float* __restrict__ Kt,
                                                const int*   __restrict__ indices,
                                                float* __restrict__ out) {
    __shared__ __align__(16) float ktab[32 * KPAD];   // 11.0 KB kernel tables
    __shared__ float obuf[32 * 33];                   // padded 32x32 result tile

    const int b0 = blockIdx.x * 32;
    const int l0 = blockIdx.y * 32;
    const int lane = threadIdx.x & 31;

    // ---- stage 32 kernel tables with CDNA5 async copy engine (b128/lane) ----
    {
        const float* src = Kt + (size_t)l0 * KPAD;    // contiguous 32*KPAD floats
        const int n128 = (32 * KPAD) / 4;             // 704 16-byte chunks
#pragma unroll 1
        for (int e = threadIdx.x; e < n128; e += 256) {
            uint32_t loff = (uint32_t)(uintptr_t)(ktab + e * 4); // LDS byte offset
            async_g2l_b128(loff, src + e * 4);
        }
        wait_asynccnt0();
        __syncthreads();
    }

    // ---- compute: each wave handles one lattice per pass, lanes over b -------
#pragma unroll
    for (int pass = 0; pass < 4; ++pass) {
        const int l_sub = __builtin_amdgcn_readfirstlane((int)(threadIdx.x >> 5) + 8 * pass);
        const int l = l0 + l_sub;
        const int* ip = indices + 4 * l;              // uniform -> scalar loads
        const int i0 = ip[0], i1 = ip[1], i2 = ip[2], i3 = ip[3];
        const int b = b0 + lane;

        // coalesced gathers of pre-clipped features
        const float xg0 = xT[(size_t)i0 * BATCH + b];
        const float xg1 = xT[(size_t)i1 * BATCH + b];
        const float xg2 = xT[(size_t)i2 * BATCH + b];
        const float xg3 = xT[(size_t)i3 * BATCH + b];

        // cell + fraction per rank (only 2 of 3 hat weights are non-zero)
        int c0 = (int)xg0; c0 = c0 > 1 ? 1 : c0; const float t0 = xg0 - (float)c0;
        int c1 = (int)xg1; c1 = c1 > 1 ? 1 : c1; const float t1 = xg1 - (float)c1;
        int c2 = (int)xg2; c2 = c2 > 1 ? 1 : c2; const float t2 = xg2 - (float)c2;
        int c3 = (int)xg3; c3 = c3 > 1 ? 1 : c3; const float t3 = xg3 - (float)c3;

        const int base = c0 + 3 * c1 + 9 * c2 + 27 * c3;   // <= 40
        const float* kr = ktab + l_sub * KPAD;

        // 2^4 nested lerp: 16 LDS gathers + 15 (sub,fma) pairs
        float v3[2];
#pragma unroll
        for (int e3 = 0; e3 < 2; ++e3) {
            float v2[2];
#pragma unroll
            for (int e2 = 0; e2 < 2; ++e2) {
                float v1[2];
#pragma unroll
                for (int e1 = 0; e1 < 2; ++e1) {
                    const int o = base + e3 * 27 + e2 * 9 + e1 * 3;
                    const float k0 = kr[o];
                    const float k1 = kr[o + 1];
                    v1[e1] = fmaf(t0, k1 - k0, k0);
                }
                v2[e2] = fmaf(t1, v1[1] - v1[0], v1[0]);
            }
            v3[e3] = fmaf(t2, v2[1] - v2[0], v2[0]);
        }
        obuf[l_sub * 33 + lane] = fmaf(t3, v3[1] - v3[0], v3[0]);
    }

    __syncthreads();

    // ---- store: lanes over l -> full 128B coalesced lines of out[b][l] ------
#pragma unroll
    for (int g = 0; g < 4; ++g) {
        const int bb = (int)(threadIdx.x >> 5) + g * 8;
        const int ll = lane;
        out[(size_t)(b0 + bb) * NLAT + (l0 + ll)] = obuf[ll * 33 + bb]; // conflict-free (33 odd)
    }
}

// ---------------------------------------------------------------------------
extern "C" void kernel_launch(void* const* d_in, const int* in_sizes, int n_in,
                              void* d_out, int out_size, void* d_ws, size_t ws_size,
                              hipStream_t stream) {
    const float* x    = (const float*)d_in[0];   // (BATCH, NFEAT)
    const float* kern = (const float*)d_in[1];   // (NVERT, NLAT)
    const int*   idx  = (const int*)d_in[2];     // (NLAT, 4)
    float* out = (float*)d_out;                  // (BATCH, NLAT)

    float* xT = (float*)d_ws;                            // NFEAT*BATCH floats (1 MB)
    float* Kt = xT + (size_t)NFEAT * BATCH;              // NLAT*KPAD floats (352 KB)

    RTL_xpose_clip<<<dim3(NFEAT / 32, BATCH / 32), 256, 0, stream>>>(x, xT);
    RTL_kpose<<<dim3(NLAT / 256), 256, 0, stream>>>(kern, Kt);
    RTL_main<<<dim3(BATCH / 32, NLAT / 32), 256, 0, stream>>>(xT, Kt, idx, out);
}